// SAGEConv_Custom_13623636263497
// MI455X (gfx1250) — compile-verified
//
#include <hip/hip_runtime.h>

typedef __attribute__((ext_vector_type(2))) float v2f;
typedef __attribute__((ext_vector_type(8))) float v8f;

#define D 64  // D_IN == D_OUT == 64

// ---------------------------------------------------------------------------
// Kernel 1: zero workspace (neigh_sum [N*64] + deg [N])
// ---------------------------------------------------------------------------
__global__ void sage_zero_ws(float* __restrict__ ws, int n) {
    int i = blockIdx.x * blockDim.x + threadIdx.x;
    if (i < n) ws[i] = 0.0f;
}

// ---------------------------------------------------------------------------
// Kernel 2: edge scatter. 16 lanes per edge, float4 gather of feat[src],
// scale by edge_weight*edge_mask, f32 atomics into neigh_sum[dst]; lane 0
// bumps deg[dst]. Entire working set is L2-resident (192MB L2 >> 55MB).
// ---------------------------------------------------------------------------
__global__ void sage_scatter(const float* __restrict__ feat,
                             const int*   __restrict__ src,
                             const int*   __restrict__ dst,
                             const float* __restrict__ ew,
                             const float* __restrict__ em,
                             float* __restrict__ nsum,
                             float* __restrict__ deg,
                             int n_edges) {
    long long tid = (long long)blockIdx.x * blockDim.x + threadIdx.x;
    int e   = (int)(tid >> 4);
    int sub = (int)(tid & 15);
    if (e >= n_edges) return;
    int s = src[e];
    int d = dst[e];
    float w = ew[e] * em[e];
    const float4 v = ((const float4*)(feat + (long long)s * D))[sub];
    float* out = nsum + (long long)d * D + sub * 4;
    atomicAdd(out + 0, v.x * w);
    atomicAdd(out + 1, v.y * w);
    atomicAdd(out + 2, v.z * w);
    atomicAdd(out + 3, v.w * w);
    if (sub == 0) atomicAdd(deg + d, 1.0f);
}

// ---------------------------------------------------------------------------
// Kernel 3: fused dual GEMM via V_WMMA_F32_16X16X4_F32.
// One wave = 16-node strip. out = feat@Wself^T + (nsum/deg)@Wneigh^T + biases.
// A 16x4 f32 layout: lane L -> row M=L&15, K = 4*kk + (L>=16?2:0) + {0,1}.
// B  4x16 f32 layout: lane L -> col N=L&15, same K pattern; B[k][n]=W[n*64+k].
// C/D 16x16: VGPR r -> row r (lanes 0-15) / r+8 (lanes 16-31), N = lane&15.
// ---------------------------------------------------------------------------
__global__ void sage_wmma(const float* __restrict__ feat,
                          const float* __restrict__ nsum,
                          const float* __restrict__ deg,
                          const float* __restrict__ Wself,
                          const float* __restrict__ bself,
                          const float* __restrict__ Wneigh,
                          const float* __restrict__ bneigh,
                          float* __restrict__ out,
                          int n_nodes) {
    int wave = blockIdx.x * (blockDim.x >> 5) + (threadIdx.x >> 5);
    int nwaves = (n_nodes + 15) >> 4;
    if (wave >= nwaves) return;

    int lane = threadIdx.x & 31;
    int mrow = lane & 15;
    int koff = (lane >> 4) << 1;          // 0 for lanes 0-15, 2 for lanes 16-31
    int m    = wave * 16 + mrow;
    int mld  = m < n_nodes ? m : n_nodes - 1;   // clamp loads for tail strip

    float dscale = 1.0f / fmaxf(deg[mld], 1.0f);
    const float* frow = feat + (long long)mld * D;
    const float* hrow = nsum + (long long)mld * D;

    // Preload this strip's A-fragments once (feat row + mean-scaled neigh row).
    v2f af[16], ah[16];
#pragma unroll
    for (int kk = 0; kk < 16; ++kk) {
        int kb = kk * 4 + koff;
        af[kk].x = frow[kb];
        af[kk].y = frow[kb + 1];
        ah[kk].x = hrow[kb] * dscale;
        ah[kk].y = hrow[kb + 1] * dscale;
    }

    int nc = lane & 15;
    int rbase = wave * 16 + ((lane >> 4) << 3);   // +8 rows for upper half-wave

    for (int nt = 0; nt < 4; ++nt) {
        int n0 = nt * 16;
        float bias = bself[n0 + nc] + bneigh[n0 + nc];
        v8f c;
#pragma unroll
        for (int r = 0; r < 8; ++r) c[r] = bias;

        const float* wsrow = Wself  + (long long)(n0 + nc) * D;
        const float* wnrow = Wneigh + (long long)(n0 + nc) * D;
#pragma unroll
        for (int kk = 0; kk < 16; ++kk) {
            int kb = kk * 4 + koff;
            v2f bs, bn;
            bs.x = wsrow[kb];  bs.y = wsrow[kb + 1];
            bn.x = wnrow[kb];  bn.y = wnrow[kb + 1];
            c = __builtin_amdgcn_wmma_f32_16x16x4_f32(
                    false, af[kk], false, bs, (short)0, c, false, false);
            c = __builtin_amdgcn_wmma_f32_16x16x4_f32(
                    false, ah[kk], false, bn, (short)0, c, false, false);
        }
#pragma unroll
        for (int r = 0; r < 8; ++r) {
            int row = rbase + r;
            if (row < n_nodes)
                out[(long long)row * D + n0 + nc] = c[r];
        }
    }
}

// ---------------------------------------------------------------------------
// launcher
// ---------------------------------------------------------------------------
extern "C" void kernel_launch(void* const* d_in, const int* in_sizes, int n_in,
                              void* d_out, int out_size, void* d_ws, size_t ws_size,
                              hipStream_t stream) {
    const float* feat   = (const float*)d_in[0];
    const int*   src    = (const int*)  d_in[1];
    const int*   dst    = (const int*)  d_in[2];
    const float* ew     = (const float*)d_in[3];
    const float* em     = (const float*)d_in[4];
    const float* Wself  = (const float*)d_in[5];
    const float* bself  = (const float*)d_in[6];
    const float* Wneigh = (const float*)d_in[7];
    const float* bneigh = (const float*)d_in[8];
    float* out = (float*)d_out;

    int n_nodes = in_sizes[0] / D;
    int n_edges = in_sizes[1];

    float* nsum = (float*)d_ws;
    float* deg  = nsum + (long long)n_nodes * D;

    // 1) zero accumulators (neigh_sum + deg)
    {
        int n = n_nodes * D + n_nodes;
        int blk = 256;
        int grd = (n + blk - 1) / blk;
        sage_zero_ws<<<grd, blk, 0, stream>>>(nsum, n);
    }
    // 2) edge scatter (16 lanes / edge)
    {
        long long threads = (long long)n_edges * 16;
        int blk = 256;
        long long grd = (threads + blk - 1) / blk;
        sage_scatter<<<(int)grd, blk, 0, stream>>>(feat, src, dst, ew, em,
                                                   nsum, deg, n_edges);
    }
    // 3) WMMA fused dual-GEMM + bias + mean
    {
        int nwaves = (n_nodes + 15) / 16;           // one wave per 16-node strip
        int waves_per_block = 8;                    // 256 threads
        int grd = (nwaves + waves_per_block - 1) / waves_per_block;
        sage_wmma<<<grd, 256, 0, stream>>>(feat, nsum, deg, Wself, bself,
                                           Wneigh, bneigh, out, n_nodes);
    }
}